// GAE_Encoder_33260226740269
// MI455X (gfx1250) — compile-verified
//
#include <hip/hip_runtime.h>
#include <hip/hip_bf16.h>

typedef __attribute__((ext_vector_type(2))) float v2f;
typedef __attribute__((ext_vector_type(8))) float v8f;

// ---------------------------------------------------------------- utilities

__global__ void zero_f32(float* __restrict__ p, long long n) {
  long long i = (long long)blockIdx.x * blockDim.x + threadIdx.x;
  if (i < n) p[i] = 0.0f;
}

// deg[col[e]] += 1 for every edge (self-loop +1 added in rsqrt pass)
__global__ void deg_kernel(const long long* __restrict__ col,
                           float* __restrict__ deg, long long E) {
  long long e = (long long)blockIdx.x * blockDim.x + threadIdx.x;
  if (e < E) atomicAdd(&deg[col[e]], 1.0f);
}

// deg -> rsqrt(deg + 1)  (in place; +1 is the self loop)
__global__ void rsqrt_kernel(float* __restrict__ deg, long long n) {
  long long i = (long long)blockIdx.x * blockDim.x + threadIdx.x;
  if (i < n) deg[i] = rsqrtf(deg[i] + 1.0f);
}

// ------------------------------------------------------------- WMMA GEMM 1
// m1[N,32] = x[N,128] @ W1[128,32], fp32 WMMA 16x16x4.
// One wave per 16x16 output tile; waveId -> (mtile, ntile), ntile in {0,1}.
// Grid must be sized so every wave owns a full tile (EXEC all-1s for WMMA).
__global__ void gemm1_wmma(const float* __restrict__ x,
                           const float* __restrict__ W1,
                           float* __restrict__ m1) {
  const int wave  = (int)((blockIdx.x * blockDim.x + threadIdx.x) >> 5);
  const int lane  = threadIdx.x & 31;
  const int mtile = wave >> 1;
  const int ntile = wave & 1;
  const int l16   = lane & 15;
  const int half  = lane >> 4;           // 0: K=0,1   1: K=2,3
  const long long mbase = (long long)mtile * 16;
  const int nbase = ntile * 16;

  v8f c = {};
  const float* arow = x  + (mbase + l16) * 128 + 2 * half;   // A: row mbase+l16
  const float* bcol = W1 + (2 * half) * 32 + nbase + l16;    // B: col nbase+l16

  #pragma unroll 4
  for (int k0 = 0; k0 < 128; k0 += 4) {
    v2f a = *(const v2f*)(arow + k0);     // {A[m][k0+2h], A[m][k0+2h+1]}
    v2f b;
    b.x = bcol[k0 * 32];                  // B[k0+2h  ][n]
    b.y = bcol[k0 * 32 + 32];             // B[k0+2h+1][n]
    c = __builtin_amdgcn_wmma_f32_16x16x4_f32(
        false, a, false, b, (short)0, c, false, false);
  }

  // D layout: lanes 0-15 -> M = v, lanes 16-31 -> M = v+8; N = l16
  float* drow = m1 + (mbase + 8 * half) * 32 + nbase + l16;
  #pragma unroll
  for (int v = 0; v < 8; ++v) drow[(long long)v * 32] = c[v];
}

// ------------------------------------------------------------- WMMA GEMM 2
// m2[N,16] = h[N,32] @ W2[32,16]; one wave per 16x16 tile, K=32.
__global__ void gemm2_wmma(const float* __restrict__ h,
                           const float* __restrict__ W2,
                           float* __restrict__ m2) {
  const int wave  = (int)((blockIdx.x * blockDim.x + threadIdx.x) >> 5);
  const int lane  = threadIdx.x & 31;
  const int l16   = lane & 15;
  const int half  = lane >> 4;
  const long long mbase = (long long)wave * 16;

  v8f c = {};
  const float* arow = h  + (mbase + l16) * 32 + 2 * half;
  const float* bcol = W2 + (2 * half) * 16 + l16;

  #pragma unroll
  for (int k0 = 0; k0 < 32; k0 += 4) {
    v2f a = *(const v2f*)(arow + k0);
    v2f b;
    b.x = bcol[k0 * 16];
    b.y = bcol[k0 * 16 + 16];
    c = __builtin_amdgcn_wmma_f32_16x16x4_f32(
        false, a, false, b, (short)0, c, false, false);
  }

  float* drow = m2 + (mbase + 8 * half) * 16 + l16;
  #pragma unroll
  for (int v = 0; v < 8; ++v) drow[(long long)v * 16] = c[v];
}

// --------------------------------------------------------- edge aggregation
// One thread per (edge, channel). F = feature width (power of 2).
// Gather m[row] (contiguous F*4B per edge-group) and atomically scatter into
// agg[col] (contiguous F*4B) -> coalesced L2-resident f32 atomics.
template <int F>
__global__ void agg_kernel(const float* __restrict__ m,
                           const float* __restrict__ dinv,
                           const long long* __restrict__ row,
                           const long long* __restrict__ col,
                           float* __restrict__ agg, long long nwork) {
  long long gid = (long long)blockIdx.x * blockDim.x + threadIdx.x;
  if (gid >= nwork) return;
  long long e = gid / F;
  int ch = (int)(gid & (F - 1));
  long long r = row[e];
  long long t = col[e];
  float nrm = dinv[r] * dinv[t];
  atomicAdd(&agg[t * F + ch], m[r * F + ch] * nrm);
}

// h = relu(agg1 + m1 * dinv^2 + b1)   (self-loop folded in, in place)
__global__ void bias_relu_self(const float* __restrict__ m1,
                               const float* __restrict__ dinv,
                               const float* __restrict__ b1,
                               float* __restrict__ agg_h, long long n) {
  long long gid = (long long)blockIdx.x * blockDim.x + threadIdx.x;
  if (gid >= n) return;
  long long node = gid >> 5;
  int ch = (int)(gid & 31);
  float d = dinv[node];
  float v = agg_h[gid] + m1[gid] * d * d + b1[ch];
  agg_h[gid] = fmaxf(v, 0.0f);
}

// out += m2 * dinv^2 + b2   (self-loop + bias for layer 2)
__global__ void out_final(const float* __restrict__ m2,
                          const float* __restrict__ dinv,
                          const float* __restrict__ b2,
                          float* __restrict__ out, long long n) {
  long long gid = (long long)blockIdx.x * blockDim.x + threadIdx.x;
  if (gid >= n) return;
  long long node = gid >> 4;
  int ch = (int)(gid & 15);
  float d = dinv[node];
  out[gid] += m2[gid] * d * d + b2[ch];
}

// ------------------------------------------------------------------ driver

extern "C" void kernel_launch(void* const* d_in, const int* in_sizes, int n_in,
                              void* d_out, int out_size, void* d_ws, size_t ws_size,
                              hipStream_t stream) {
  const float*     x   = (const float*)d_in[0];
  const long long* ei  = (const long long*)d_in[1];   // int64 per reference
  const float*     W1  = (const float*)d_in[2];
  const float*     b1  = (const float*)d_in[3];
  const float*     W2  = (const float*)d_in[4];
  const float*     b2  = (const float*)d_in[5];
  float*           out = (float*)d_out;

  const long long N = in_sizes[0] / 128;   // 100000 (divisible by 16)
  const long long E = in_sizes[1] / 2;     // 6400000
  const long long* row = ei;
  const long long* col = ei + E;

  // workspace layout (256B aligned)
  auto align256 = [](size_t v) { return (v + 255) & ~(size_t)255; };
  char* w = (char*)d_ws;
  size_t off = 0;
  float* dinv = (float*)(w + off); off = align256(off + (size_t)N * 4);
  float* m1   = (float*)(w + off); off = align256(off + (size_t)N * 32 * 4);
  float* h    = (float*)(w + off); off = align256(off + (size_t)N * 32 * 4); // agg1, reused as h
  float* m2   = (float*)(w + off); off = align256(off + (size_t)N * 16 * 4);
  (void)ws_size; (void)n_in; (void)out_size;

  const int B = 256;
  // zero accumulators (poisoned workspace + graph replay -> must re-zero)
  zero_f32<<<(unsigned)((N       + B - 1) / B), B, 0, stream>>>(dinv, N);
  zero_f32<<<(unsigned)((N * 32  + B - 1) / B), B, 0, stream>>>(h, N * 32);
  zero_f32<<<(unsigned)((N * 16  + B - 1) / B), B, 0, stream>>>(out, N * 16);

  // degree + symmetric normalization
  deg_kernel  <<<(unsigned)((E + B - 1) / B), B, 0, stream>>>(col, dinv, E);
  rsqrt_kernel<<<(unsigned)((N + B - 1) / B), B, 0, stream>>>(dinv, N);

  // layer 1: WMMA transform, edge scatter, bias+self-loop+relu
  {
    long long waves = (N / 16) * 2;              // 12500 tiles
    gemm1_wmma<<<(unsigned)(waves * 32 / 128), 128, 0, stream>>>(x, W1, m1);
  }
  agg_kernel<32><<<(unsigned)((E * 32 + B - 1) / B), B, 0, stream>>>(
      m1, dinv, row, col, h, E * 32);
  bias_relu_self<<<(unsigned)((N * 32 + B - 1) / B), B, 0, stream>>>(
      m1, dinv, b1, h, N * 32);

  // layer 2: WMMA transform, edge scatter into d_out, bias+self-loop
  {
    long long waves = N / 16;                    // 6250 tiles
    gemm2_wmma<<<(unsigned)(waves * 32 / 64), 64, 0, stream>>>(h, W2, m2);
  }
  agg_kernel<16><<<(unsigned)((E * 16 + B - 1) / B), B, 0, stream>>>(
      m2, dinv, row, col, out, E * 16);
  out_final<<<(unsigned)((N * 16 + B - 1) / B), B, 0, stream>>>(
      m2, dinv, b2, out, N * 16);
}